// EquilibriumClass_32427003085424
// MI455X (gfx1250) — compile-verified
//
#include <hip/hip_runtime.h>
#include <stdint.h>

#define N_NODES   32770          // H + 2
#define H_HINGES  32768
#define N_EDGES   32769          // H + 1
#define NTHREADS  1024
#define NODES_PER 32
#define DT_C      0.001f
#define GAMMA_C   2.0f
#define LDS_BYTES (N_NODES * 2 * 4)   // 262160 B, fits in 320 KB WGP LDS

// Whole chain position state lives in LDS (dynamic shared).
extern __shared__ float sm[];

#if defined(__has_builtin) && __has_builtin(__builtin_amdgcn_s_wait_asynccnt)
#define ASYNC_WAIT0() __builtin_amdgcn_s_wait_asynccnt(0)
#else
#define ASYNC_WAIT0() asm volatile("s_wait_asynccnt 0x0" ::: "memory")
#endif

__device__ __forceinline__ float2 ldsPos(int i) {
    return *(const float2*)(sm + 2 * i);
}

// Hinge j: v1 = edge j, v2 = edge j+1.  theta = atan2(cross, dot).
// Effective stiffness K = sum over 4 buckle slots of (stiff? k_stiff : k_soft);
// the mask is piecewise-constant so jax.grad treats it as constant.
// P = g * dtheta/dv1, Q = g * dtheta/dv2 with g = K*(theta - theta_ss)/r2 folded.
__device__ __forceinline__ void hingePQ(
    float v1x, float v1y, float v2x, float v2y, int j,
    const int* __restrict__ buckle, const float* __restrict__ thetas_ss,
    float kstiff, float ksoft,
    float& Px, float& Py, float& Qx, float& Qy)
{
    float c  = v1x * v2y - v1y * v2x;
    float d  = v1x * v2x + v1y * v2y;
    float r2 = c * c + d * d;
    float inv = 1.0f / r2;
    float theta = atan2f(c, d);
    float th = thetas_ss[j];
    const int4 b = *(const int4*)(buckle + 4 * j);
    float K = 0.0f;
    K += ((b.x == 1 && theta > -th) || (b.x == -1 && theta < th)) ? kstiff : ksoft;
    K += ((b.y == 1 && theta > -th) || (b.y == -1 && theta < th)) ? kstiff : ksoft;
    K += ((b.z == 1 && theta > -th) || (b.z == -1 && theta < th)) ? kstiff : ksoft;
    K += ((b.w == 1 && theta > -th) || (b.w == -1 && theta < th)) ? kstiff : ksoft;
    float g = K * (theta - th) * inv;
    Px = g * ( d * v2y - c * v2x);
    Py = g * (-d * v2x - c * v2y);
    Qx = g * (-d * v1y - c * v1x);
    Qy = g * ( d * v1x - c * v1y);
}

__global__ __launch_bounds__(NTHREADS) void chain_relax_persistent(
    const float* __restrict__ init_pos,     // (N,2)
    const int*   __restrict__ buckle,       // (H,4)
    const float* __restrict__ thetas_ss,    // (H,)
    const float* __restrict__ rest_lengths, // (H+1,)
    const float* __restrict__ k_stiff_p,
    const float* __restrict__ k_soft_p,
    const float* __restrict__ k_stretch_p,
    const int*   __restrict__ n_steps_p,
    float*       __restrict__ out,          // [x_final (2N)] ++ [traj (nsteps*2N)]
    float*       __restrict__ vel)          // workspace: 2N floats, thread-private
{
    const int tid = threadIdx.x;
    const int s = tid * NODES_PER;
    const int e = (tid == NTHREADS - 1) ? N_NODES : (s + NODES_PER);

    const float kstiff   = k_stiff_p[0];
    const float ksoft    = k_soft_p[0];
    const float kstretch = k_stretch_p[0];
    const int   nsteps   = n_steps_p[0];

    // Zero my velocity coords (workspace is poisoned by the harness).
    for (int i = s; i < e; ++i) { vel[2*i] = 0.0f; vel[2*i + 1] = 0.0f; }

    // LDS byte address of my block (generic LDS pointer low 32 bits == LDS offset).
    const unsigned lbase = (unsigned)(uintptr_t)sm + (unsigned)s * 8u;
    const int nb = (e - s) * 8;   // bytes in my block (256 or 272 -> 16B multiple)

    // --- Async DMA: init_pos -> LDS (ASYNCcnt path) ---
    {
        unsigned long long g = (unsigned long long)(uintptr_t)init_pos
                             + (unsigned long long)s * 8ull;
        for (int o = 0; o < nb; o += 16) {
            asm volatile("global_load_async_to_lds_b128 %0, %1, off"
                         :: "v"(lbase + (unsigned)o),
                            "v"(g + (unsigned long long)o) : "memory");
        }
    }
    ASYNC_WAIT0();
    __syncthreads();

    float* traj = out + 2 * N_NODES;

    #pragma unroll 1
    for (int t = 0; t < nsteps; ++t) {
        // ---- snapshot halo (old state) before anyone writes ----
        float2 pm2 = make_float2(0.f, 0.f), pm1 = make_float2(0.f, 0.f);
        float2 he  = make_float2(0.f, 0.f), he1 = make_float2(0.f, 0.f);
        if (s >= 2)               pm2 = ldsPos(s - 2);
        if (s >= 1)               pm1 = ldsPos(s - 1);
        if (e     <= N_NODES - 1) he  = ldsPos(e);
        if (e + 1 <= N_NODES - 1) he1 = ldsPos(e + 1);
        __syncthreads();          // all snapshots taken
        ASYNC_WAIT0();            // my slice(t-1) DMA done before I overwrite my pos

        // ---- prologue: P_{s-1}, Q_{s-1}, Q_{s-2}, S_{s-1} ----
        float2 p0 = ldsPos(s);
        float2 p1 = ldsPos(s + 1);
        float Ppx = 0.f, Ppy = 0.f, Qpx = 0.f, Qpy = 0.f;
        float Qp2x = 0.f, Qp2y = 0.f, Spx = 0.f, Spy = 0.f;
        if (s >= 1) {
            float vbx = p0.x - pm1.x, vby = p0.y - pm1.y;   // edge s-1
            float v0x = p1.x - p0.x,  v0y = p1.y - p0.y;    // edge s
            hingePQ(vbx, vby, v0x, v0y, s - 1, buckle, thetas_ss,
                    kstiff, ksoft, Ppx, Ppy, Qpx, Qpy);
            float len  = sqrtf(vbx * vbx + vby * vby);
            float coef = kstretch * (len - rest_lengths[s - 1]) / len;
            Spx = coef * vbx; Spy = coef * vby;
            if (s >= 2) {
                float vax = pm1.x - pm2.x, vay = pm1.y - pm2.y;  // edge s-2
                float dx, dy;
                hingePQ(vax, vay, vbx, vby, s - 2, buckle, thetas_ss,
                        kstiff, ksoft, dx, dy, Qp2x, Qp2y);
            }
        }

        // ---- rolling per-node force + integrate (one hinge + one edge per node) ----
        float2 pcur = p0, pnext = p1;
        float vcx = p1.x - p0.x, vcy = p1.y - p0.y;   // edge i (i = s)
        #pragma unroll 1
        for (int i = s; i < e; ++i) {
            float vnx = 0.f, vny = 0.f;
            float2 p2 = make_float2(0.f, 0.f);
            if (i + 2 <= N_NODES - 1) {
                p2 = (i + 2 == e) ? he : ((i + 2 == e + 1) ? he1 : ldsPos(i + 2));
                vnx = p2.x - pnext.x; vny = p2.y - pnext.y;   // edge i+1
            }
            float Px = 0.f, Py = 0.f, Qx = 0.f, Qy = 0.f;
            if (i <= H_HINGES - 1)
                hingePQ(vcx, vcy, vnx, vny, i, buckle, thetas_ss,
                        kstiff, ksoft, Px, Py, Qx, Qy);
            float Sx = 0.f, Sy = 0.f;
            if (i <= N_EDGES - 1) {
                float len  = sqrtf(vcx * vcx + vcy * vcy);
                float coef = kstretch * (len - rest_lengths[i]) / len;
                Sx = coef * vcx; Sy = coef * vcy;
            }
            // f_i = P_i - P_{i-1} + Q_{i-1} - Q_{i-2} + S_i - S_{i-1}
            float fx = Px - Ppx + Qpx - Qp2x + Sx - Spx;
            float fy = Py - Ppy + Qpy - Qp2y + Sy - Spy;

            float vx = vel[2*i], vy = vel[2*i + 1];
            float nvx = vx + DT_C * (fx - GAMMA_C * vx);
            float nvy = vy + DT_C * (fy - GAMMA_C * vy);
            float nx  = pcur.x + DT_C * nvx;
            float ny  = pcur.y + DT_C * nvy;
            if (i < 2) { nvx = 0.f; nvy = 0.f; nx = pcur.x; ny = pcur.y; } // pinned
            vel[2*i] = nvx; vel[2*i + 1] = nvy;
            *(float2*)(sm + 2 * i) = make_float2(nx, ny);

            // rotate rolling state
            Ppx = Px;  Ppy = Py;
            Qp2x = Qpx; Qp2y = Qpy; Qpx = Qx; Qpy = Qy;
            Spx = Sx;  Spy = Sy;
            pcur = pnext; pnext = p2; vcx = vnx; vcy = vny;
        }
        __syncthreads();   // state(t+1) complete in LDS

        // ---- Async DMA: stream slice t (LDS -> HBM), non-temporal so the
        //      262 MB trajectory stream never evicts hot L2 state; overlaps
        //      the next step's force math. ----
        asm volatile("s_wait_dscnt 0x0" ::: "memory");  // my LDS writes committed
        {
            unsigned long long g = (unsigned long long)(uintptr_t)traj
                                 + (unsigned long long)t * (unsigned long long)(2 * N_NODES * 4)
                                 + (unsigned long long)s * 8ull;
            for (int o = 0; o < nb; o += 16) {
                asm volatile("global_store_async_from_lds_b128 %0, %1, off th:TH_STORE_NT"
                             :: "v"(g + (unsigned long long)o),
                                "v"(lbase + (unsigned)o) : "memory");
            }
        }
    }

    // ---- x_final = final LDS state -> out[0 .. 2N) ----
    {
        unsigned long long g = (unsigned long long)(uintptr_t)out
                             + (unsigned long long)s * 8ull;
        for (int o = 0; o < nb; o += 16) {
            asm volatile("global_store_async_from_lds_b128 %0, %1, off th:TH_STORE_NT"
                         :: "v"(g + (unsigned long long)o),
                            "v"(lbase + (unsigned)o) : "memory");
        }
    }
    ASYNC_WAIT0();
}

extern "C" void kernel_launch(void* const* d_in, const int* in_sizes, int n_in,
                              void* d_out, int out_size, void* d_ws, size_t ws_size,
                              hipStream_t stream) {
    (void)in_sizes; (void)n_in; (void)out_size; (void)ws_size;
    // Opt in to >64KB dynamic LDS (CDNA5 WGP has 320KB). Idempotent; error ignored.
    (void)hipFuncSetAttribute((const void*)chain_relax_persistent,
                              hipFuncAttributeMaxDynamicSharedMemorySize, LDS_BYTES);
    chain_relax_persistent<<<1, NTHREADS, LDS_BYTES, stream>>>(
        (const float*)d_in[0],   // init_pos
        (const int*)  d_in[1],   // buckle
        (const float*)d_in[2],   // thetas_ss
        (const float*)d_in[3],   // rest_lengths
        (const float*)d_in[4],   // k_stiff
        (const float*)d_in[5],   // k_soft
        (const float*)d_in[6],   // k_stretch
        (const int*)  d_in[7],   // n_steps
        (float*)d_out,
        (float*)d_ws);           // velocities (2N floats)
}